// HybridKPlanesEncoder_89215060672781
// MI455X (gfx1250) — compile-verified
//
#include <hip/hip_runtime.h>
#include <math.h>
#include <stdint.h>

#define NLEVELS   16
#define HASH_MASK 0x7FFFFu        // all hashed levels have size == 2^19 exactly
#define KP_PRIME  2654435761u

typedef float v2f __attribute__((ext_vector_type(2)));
typedef int   v4i __attribute__((vector_size(16)));

// AS-qualified pointers (AS1 = global, AS3 = LDS)
typedef __attribute__((address_space(1))) v4i*        gv4i_p;
typedef __attribute__((address_space(3))) v4i*        lv4i_p;
typedef __attribute__((address_space(3))) const float* lds_cfp;

struct LevelMeta { int res[NLEVELS]; int off[NLEVELS]; };

// ---- gfx1250 async global->LDS path (guarded so compile never breaks) ----
#if defined(__has_builtin)
#  if __has_builtin(__builtin_amdgcn_global_load_async_to_lds_b128) && \
      __has_builtin(__builtin_amdgcn_s_wait_asynccnt)
#    define KP_HAVE_ASYNC 1
#  endif
#endif
#ifndef KP_HAVE_ASYNC
#  define KP_HAVE_ASYNC 0
#endif

// Per-axis bilinear setup, computed once per coordinate per level and shared
// across the planes that reuse that coordinate.
struct Axis { int i0, i1; float w0, w1; };

__device__ __forceinline__ Axis axis_setup(float c, int res)
{
  Axis a;
  const float p = c * (float)(res - 1);
  const float f = floorf(p);
  a.w1 = p - f;
  a.w0 = 1.0f - a.w1;
  a.i0 = (int)f;
  a.i1 = min(a.i0 + 1, res - 1);
  return a;
}

__device__ __forceinline__ void plane_accum(const v2f* __restrict__ tab,
                                            const Axis& ax, const Axis& ay,
                                            int res, int off, bool use_hash,
                                            float& a0, float& a1)
{
  int i00, i01, i10, i11;
  if (use_hash) {                       // size == 2^19 -> modulo is a mask
    const unsigned hy0 = (unsigned)ay.i0 * KP_PRIME;
    const unsigned hy1 = (unsigned)ay.i1 * KP_PRIME;
    i00 = (int)(((unsigned)ax.i0 ^ hy0) & HASH_MASK);
    i01 = (int)(((unsigned)ax.i0 ^ hy1) & HASH_MASK);
    i10 = (int)(((unsigned)ax.i1 ^ hy0) & HASH_MASK);
    i11 = (int)(((unsigned)ax.i1 ^ hy1) & HASH_MASK);
  } else {                              // dense levels: coherent indexing
    i00 = ax.i0 * res + ay.i0;
    i01 = ax.i0 * res + ay.i1;
    i10 = ax.i1 * res + ay.i0;
    i11 = ax.i1 * res + ay.i1;
  }

  const v2f f00 = tab[off + i00];       // global_load_b64 gathers (L2-resident)
  const v2f f01 = tab[off + i01];
  const v2f f10 = tab[off + i10];
  const v2f f11 = tab[off + i11];

  const float w00 = ax.w0 * ay.w0, w01 = ax.w0 * ay.w1;
  const float w10 = ax.w1 * ay.w0, w11 = ax.w1 * ay.w1;
  a0 = fmaf(w11, f11.x, fmaf(w10, f10.x, fmaf(w01, f01.x, fmaf(w00, f00.x, a0))));
  a1 = fmaf(w11, f11.y, fmaf(w10, f10.y, fmaf(w01, f01.y, fmaf(w00, f00.y, a1))));
}

__global__ __launch_bounds__(256) void kplanes_encode_kernel(
    const float* __restrict__ x,
    const float* __restrict__ t0,
    const float* __restrict__ t1,
    const float* __restrict__ t2,
    float* __restrict__ out,
    LevelMeta meta, int N)
{
  const int tid = (int)threadIdx.x;
  const int n   = (int)blockIdx.x * 256 + tid;

  float xv0, xv1, xv2;

#if KP_HAVE_ASYNC
  // Stage the streaming coordinate reads through LDS with the CDNA5 async
  // DMA path. Each wave stages only its own 32 points (384 B = 24 x b128),
  // so a single per-wave s_wait_asynccnt(0) is sufficient -- no barrier.
  __shared__ __align__(16) unsigned char smem[256 * 12];
  const int    wave     = tid >> 5;
  const int    lane     = tid & 31;
  const size_t blk_byte = (size_t)blockIdx.x * (256u * 12u);
  const bool   full     = (blk_byte + 256u * 12u) <= (size_t)N * 12u;
  if (full) {
    if (lane < 24) {
      const unsigned char* g = (const unsigned char*)x + blk_byte
                             + (size_t)wave * 384u + (size_t)lane * 16u;
      unsigned char* l = smem + wave * 384 + lane * 16;
      __builtin_amdgcn_global_load_async_to_lds_b128(
          (gv4i_p)(uintptr_t)g,
          (lv4i_p)(unsigned int)(uintptr_t)l,
          0, 0);
    }
    __builtin_amdgcn_s_wait_asynccnt(0);
    // Read back via an explicit AS3 pointer so this is guaranteed to lower
    // to ds_load_b32 (DScnt only) rather than a flat access.
    const lds_cfp sx = (lds_cfp)(unsigned int)(uintptr_t)
                       (smem + wave * 384 + lane * 12);
    xv0 = sx[0]; xv1 = sx[1]; xv2 = sx[2];
  } else {
    if (n >= N) return;
    xv0 = x[(size_t)n * 3 + 0];
    xv1 = x[(size_t)n * 3 + 1];
    xv2 = x[(size_t)n * 3 + 2];
  }
#else
  if (n >= N) return;
  xv0 = x[(size_t)n * 3 + 0];
  xv1 = x[(size_t)n * 3 + 1];
  xv2 = x[(size_t)n * 3 + 2];
#endif
  if (n >= N) return;

  // jnp.clip(coords, 0, 1)
  xv0 = fminf(fmaxf(xv0, 0.0f), 1.0f);
  xv1 = fminf(fmaxf(xv1, 0.0f), 1.0f);
  xv2 = fminf(fmaxf(xv2, 0.0f), 1.0f);

  const v2f* __restrict__ tb0 = (const v2f*)t0;
  const v2f* __restrict__ tb1 = (const v2f*)t1;
  const v2f* __restrict__ tb2 = (const v2f*)t2;
  v2f* __restrict__ orow = (v2f*)(out + (size_t)n * 32);

  // Unroll x4: ~48 independent b64 gathers in flight per wave (fits the
  // 6-bit LOADcnt), and adjacent NT b64 stores can merge into NT b128.
#pragma unroll 4
  for (int l = 0; l < NLEVELS; ++l) {
    const int  res = meta.res[l];
    const int  off = meta.off[l];
    const bool uh  = res >= 725;        // res^2 > 2^19  <=>  res >= 725

    // One axis setup per coordinate; planes share them:
    //   plane(0,1): x-axis = c0, y-axis = c1
    //   plane(0,2): x-axis = c0, y-axis = c2
    //   plane(1,2): x-axis = c1, y-axis = c2
    const Axis a0x = axis_setup(xv0, res);
    const Axis a1x = axis_setup(xv1, res);
    const Axis a2x = axis_setup(xv2, res);

    float s0 = 0.0f, s1 = 0.0f;
    plane_accum(tb0, a0x, a1x, res, off, uh, s0, s1);
    plane_accum(tb1, a0x, a2x, res, off, uh, s0, s1);
    plane_accum(tb2, a1x, a2x, res, off, uh, s0, s1);

    v2f v; v.x = s0; v.y = s1;
    // Non-temporal: keep the ~66 MB tables resident in the 192 MB L2;
    // the 128 MB output is write-once streaming.
    __builtin_nontemporal_store(v, orow + l);
  }
}

static void compute_meta(LevelMeta& m)
{
  // Exact replica of the reference _level_meta() in double precision.
  // (Do NOT hardcode: e.g. level 12 = 776.047 -> 777, level 14 = 1482.0037
  //  -> 1483; every value sits >=0.004 from an integer boundary, so the
  //  host double computation matches NumPy robustly.)
  const double b = exp(log(2048.0 / 16.0) / 15.0);
  int off = 0;
  for (int l = 0; l < NLEVELS; ++l) {
    const int res = (int)ceil(16.0 * pow(b, (double)l));
    long long n = (long long)res * (long long)res;
    if (n > 524288LL) n = 524288LL;
    n = ((n + 7) / 8) * 8;
    m.res[l] = res;
    m.off[l] = off;
    off += (int)n;
  }
}

extern "C" void kernel_launch(void* const* d_in, const int* in_sizes, int n_in,
                              void* d_out, int out_size, void* d_ws, size_t ws_size,
                              hipStream_t stream)
{
  const float* x  = (const float*)d_in[0];
  const float* t0 = (const float*)d_in[1];
  const float* t1 = (const float*)d_in[2];
  const float* t2 = (const float*)d_in[3];
  float* out = (float*)d_out;

  const int N = in_sizes[0] / 3;

  LevelMeta meta;
  compute_meta(meta);

  const int blocks = (N + 255) / 256;
  hipLaunchKernelGGL(kplanes_encode_kernel, dim3(blocks), dim3(256), 0, stream,
                     x, t0, t1, t2, out, meta, N);
}